// OuterProduct_23218593202746
// MI455X (gfx1250) — compile-verified
//
#include <hip/hip_runtime.h>

typedef __attribute__((ext_vector_type(2))) float v2f;
typedef __attribute__((ext_vector_type(8))) float v8f;

#define A_N 256
#define F_N 256
#define C_N 32
#define LN_EPS 1e-5f

// ---------------------------------------------------------------------------
// Kernel 1: LayerNorm + dual projection. One block per node row.
//   L[a,c] = (LN(x[a,:]) @ Wl + bl)[c] * mask[a]
//   R[a,c] = (LN(x[a,:]) @ Wr + br)[c] * mask[a]
// ---------------------------------------------------------------------------
__global__ __launch_bounds__(256) void ln_proj_kernel(
    const float* __restrict__ x, const float* __restrict__ mask,
    const float* __restrict__ gamma, const float* __restrict__ beta,
    const float* __restrict__ Wl, const float* __restrict__ bl,
    const float* __restrict__ Wr, const float* __restrict__ br,
    float* __restrict__ Lout, float* __restrict__ Rout)
{
    __shared__ float s1[256];
    __shared__ float s2[256];
    __shared__ float act[256];

    const int a = blockIdx.x;
    const int t = threadIdx.x;

    const float v = x[a * F_N + t];
    s1[t] = v;
    s2[t] = v * v;
    __syncthreads();
    for (int off = 128; off > 0; off >>= 1) {
        if (t < off) { s1[t] += s1[t + off]; s2[t] += s2[t + off]; }
        __syncthreads();
    }
    const float mu  = s1[0] * (1.0f / F_N);
    const float var = s2[0] * (1.0f / F_N) - mu * mu;
    const float inv = rsqrtf(var + LN_EPS);

    act[t] = (v - mu) * inv * gamma[t] + beta[t];
    __syncthreads();   // also fences the s1[0]/s2[0] reads above

    // 64 outputs (32 L + 32 R); 4 threads cooperate per output.
    const int c   = t >> 2;
    const int sub = t & 3;
    const int cc  = c & (C_N - 1);
    const float* __restrict__ W = (c < C_N) ? Wl : Wr;
    const int f0 = sub * 64;
    float p = 0.0f;
#pragma unroll 8
    for (int f = 0; f < 64; ++f)
        p = fmaf(act[f0 + f], W[(f0 + f) * C_N + cc], p);
    s1[t] = p;
    __syncthreads();
    if (sub == 0) {
        const float s = s1[t] + s1[t + 1] + s1[t + 2] + s1[t + 3];
        const float m = mask[a];
        if (c < C_N) Lout[a * C_N + cc] = (s + bl[cc]) * m;
        else         Rout[a * C_N + cc] = (s + br[cc]) * m;
    }
}

// ---------------------------------------------------------------------------
// Kernel 2: one block per 'a'.
//   Stage R[256x32] (padded) and T_a[d][f] = sum_c L[a,c]*Wo[(c*32+d)*256+f]
//   in LDS, then out[a] = R @ T_a + bo via V_WMMA_F32_16X16X4_F32.
//   Store-bandwidth bound: 67 MB fp32 output -> ~2.9 us floor @ 23.3 TB/s.
// ---------------------------------------------------------------------------
#define R_STRIDE (C_N + 1)   // pad to break LDS bank conflicts on A-frag reads

__global__ __launch_bounds__(256) void outer_wmma_kernel(
    const float* __restrict__ Lmat, const float* __restrict__ Rmat,
    const float* __restrict__ Wo, const float* __restrict__ bo,
    float* __restrict__ out)
{
    __shared__ float Lsh[C_N];
    __shared__ float Tsh[C_N * F_N];        // [d][f]  32 KB
    __shared__ float Rsh[A_N * R_STRIDE];   // [b][d]  ~33 KB (padded)
    __shared__ float bosh[F_N];

    const int a = blockIdx.x;
    const int t = threadIdx.x;

    if (t < C_N) Lsh[t] = Lmat[a * C_N + t];
    bosh[t] = bo[t];
    for (int i = t; i < A_N * C_N; i += 256)
        Rsh[(i >> 5) * R_STRIDE + (i & 31)] = Rmat[i];
    __syncthreads();

    // T_a[d][f], thread owns column f = t (coalesced Wo reads, L2-resident).
    {
        const int f = t;
        for (int d = 0; d < C_N; ++d) {
            float s = 0.0f;
#pragma unroll
            for (int c = 0; c < C_N; ++c)
                s = fmaf(Lsh[c], Wo[((c * C_N + d) << 8) + f], s);
            Tsh[d * F_N + f] = s;
        }
    }
    __syncthreads();

    const int wave = t >> 5;
    const int lane = t & 31;
    const int hf   = lane >> 4;   // half-wave selects K sub-pair / M+8
    const int lid  = lane & 15;

    // 256 16x16 output tiles over (b, f); 8 waves, 32 tiles each.
    for (int tile = wave; tile < 256; tile += 8) {
        const int bt = tile >> 4;
        const int ft = tile & 15;

        v8f acc = {0.f, 0.f, 0.f, 0.f, 0.f, 0.f, 0.f, 0.f};
        const float* __restrict__ rrow = &Rsh[(bt * 16 + lid) * R_STRIDE];
        const float* __restrict__ tcol = &Tsh[ft * 16 + lid];

#pragma unroll
        for (int kk = 0; kk < 8; ++kk) {
            const int k0 = kk * 4 + hf * 2;
            v2f afrag, bfrag;
            // A 16x4: lane -> M=lid, regs hold K = k0, k0+1 (half selects 0/2)
            afrag.x = rrow[k0];
            afrag.y = rrow[k0 + 1];
            // B 4x16: lane -> N=lid, regs hold K = k0, k0+1
            bfrag.x = tcol[(size_t)k0 * F_N];
            bfrag.y = tcol[(size_t)(k0 + 1) * F_N];
            acc = __builtin_amdgcn_wmma_f32_16x16x4_f32(
                false, afrag, false, bfrag, (short)0, acc, false, false);
        }

        // C/D layout: VGPR r -> M = r + 8*hf, N = lid
        const float bias = bosh[ft * 16 + lid];
        float* __restrict__ orow =
            out + ((size_t)a * A_N + (size_t)(bt * 16 + hf * 8)) * F_N + ft * 16 + lid;
#pragma unroll
        for (int r = 0; r < 8; ++r)
            orow[(size_t)r * F_N] = acc[r] + bias;
    }
}

// ---------------------------------------------------------------------------
extern "C" void kernel_launch(void* const* d_in, const int* in_sizes, int n_in,
                              void* d_out, int out_size, void* d_ws, size_t ws_size,
                              hipStream_t stream) {
    const float* node_vec  = (const float*)d_in[0];
    const float* node_mask = (const float*)d_in[1];
    const float* ln_scale  = (const float*)d_in[2];
    const float* ln_bias   = (const float*)d_in[3];
    const float* Wl        = (const float*)d_in[4];
    const float* bl        = (const float*)d_in[5];
    const float* Wr        = (const float*)d_in[6];
    const float* br        = (const float*)d_in[7];
    const float* Wo        = (const float*)d_in[8];
    const float* bo        = (const float*)d_in[9];

    float* out  = (float*)d_out;
    float* Lbuf = (float*)d_ws;                 // 32 KB
    float* Rbuf = Lbuf + A_N * C_N;             // 32 KB

    ln_proj_kernel<<<A_N, 256, 0, stream>>>(node_vec, node_mask, ln_scale, ln_bias,
                                            Wl, bl, Wr, br, Lbuf, Rbuf);
    outer_wmma_kernel<<<A_N, 256, 0, stream>>>(Lbuf, Rbuf, Wo, bo, out);
}